// GATLayerDecomposed_51994874085808
// MI455X (gfx1250) — compile-verified
//
#include <hip/hip_runtime.h>
#include <math.h>

typedef __attribute__((ext_vector_type(2))) float v2f;
typedef __attribute__((ext_vector_type(8))) float v8f;

#define IN_CH 128
#define OUT_CH 64
#define NEG_SLOPE 0.2f

// ---------- float <-> order-preserving uint (for atomic max on floats) ----------
__device__ __forceinline__ unsigned f2ord(float f) {
  unsigned u = __float_as_uint(f);
  return (u & 0x80000000u) ? ~u : (u | 0x80000000u);
}
__device__ __forceinline__ float ord2f(unsigned u) {
  return (u & 0x80000000u) ? __uint_as_float(u ^ 0x80000000u) : __uint_as_float(~u);
}

// ---------- Kernel 1: Wx = x @ W via V_WMMA_F32_16X16X4_F32 ----------
// One wave32 computes a 16x16 output tile. M-tiles over nodes, 4 N-tiles (OUT_CH=64).
// A (16x4 f32): lanes 0-15 hold M=lane, K={k0,k0+1}; lanes 16-31 hold M=lane-16, K={k0+2,k0+3}.
// B (4x16 f32): VGPR0 = rows K=k0 (lanes 0-15) / K=k0+2 (lanes 16-31); VGPR1 = K=k0+1 / k0+3.
__global__ void gemm_xW(const float* __restrict__ x, const float* __restrict__ W,
                        float* __restrict__ Wx, int nNodes, int numMTiles) {
  int waveId = (blockIdx.x * blockDim.x + threadIdx.x) >> 5;
  int lane   = threadIdx.x & 31;
  int ntile  = waveId & 3;
  int mtile  = waveId >> 2;
  if (mtile >= numMTiles) return;

  int half = lane >> 4;   // 0: lanes 0-15, 1: lanes 16-31
  int l15  = lane & 15;

  int arow  = mtile * 16 + l15;
  int arowc = (arow < nNodes) ? arow : (nNodes - 1);
  const float* aptr = x + (size_t)arowc * IN_CH + half * 2;          // K base for this lane half
  int bcol = ntile * 16 + l15;
  const float* bptr = W + (size_t)(half * 2) * OUT_CH + bcol;

  v8f acc = {};
#pragma unroll
  for (int k0 = 0; k0 < IN_CH; k0 += 4) {
    float2 av = *(const float2*)(aptr + k0);                          // 8B aligned
    v2f a; a.x = av.x; a.y = av.y;
    v2f b;
    b.x = bptr[(size_t)(k0 + 0) * OUT_CH];
    b.y = bptr[(size_t)(k0 + 1) * OUT_CH];
    acc = __builtin_amdgcn_wmma_f32_16x16x4_f32(false, a, false, b, (short)0, acc, false, false);
  }

  // D layout: VGPR v -> row mtile*16 + v + half*8, col = ntile*16 + l15
#pragma unroll
  for (int v = 0; v < 8; ++v) {
    int row = mtile * 16 + half * 8 + v;
    if (row < nNodes) Wx[(size_t)row * OUT_CH + ntile * 16 + l15] = acc[v];
  }
}

// ---------- Kernel 2: init output + segment buffers ----------
__global__ void init_buffers(float* __restrict__ hout, unsigned* __restrict__ emax_u,
                             float* __restrict__ sum_exp, int nNodes, int outElems) {
  int i = blockIdx.x * blockDim.x + threadIdx.x;
  if (i < outElems) hout[i] = 0.0f;
  if (i < nNodes) {
    emax_u[i]  = f2ord(-1e9f);   // matches reference's -1e9 floor for empty segments
    sum_exp[i] = 0.0f;
  }
}

// ---------- Kernel 3: per-node attention scalars ----------
__global__ void node_scores(const float* __restrict__ Wx, const float* __restrict__ att_src,
                            const float* __restrict__ att_dst, float* __restrict__ s_src,
                            float* __restrict__ s_dst, int nNodes) {
  __shared__ float sa[OUT_CH];
  __shared__ float sd[OUT_CH];
  if (threadIdx.x < OUT_CH) {
    sa[threadIdx.x] = att_src[threadIdx.x];
    sd[threadIdx.x] = att_dst[threadIdx.x];
  }
  __syncthreads();
  int n = blockIdx.x * blockDim.x + threadIdx.x;
  if (n >= nNodes) return;
  const float4* wp = (const float4*)(Wx + (size_t)n * OUT_CH);
  float ss = 0.0f, dd = 0.0f;
#pragma unroll
  for (int q = 0; q < OUT_CH / 4; ++q) {
    float4 w = wp[q];
    ss += w.x * sa[4 * q + 0] + w.y * sa[4 * q + 1] + w.z * sa[4 * q + 2] + w.w * sa[4 * q + 3];
    dd += w.x * sd[4 * q + 0] + w.y * sd[4 * q + 1] + w.z * sd[4 * q + 2] + w.w * sd[4 * q + 3];
  }
  s_src[n] = ss;
  s_dst[n] = dd;
}

// ---------- Kernel 4: edge score + LeakyReLU + segment max ----------
__global__ void edge_scores_max(const int* __restrict__ ei, const float* __restrict__ s_src,
                                const float* __restrict__ s_dst, float* __restrict__ e_val,
                                unsigned* __restrict__ emax_u, int E) {
  int e = blockIdx.x * blockDim.x + threadIdx.x;
  if (e >= E) return;
  int s = ei[e];
  int d = ei[E + e];
  float sc = s_dst[d] + s_src[s];
  sc = (sc >= 0.0f) ? sc : NEG_SLOPE * sc;
  e_val[e] = sc;
  atomicMax(&emax_u[d], f2ord(sc));
}

// ---------- Kernel 5: exp(e - max) + segment sum ----------
__global__ void edge_exp_sum(const int* __restrict__ ei, float* __restrict__ e_val,
                             const unsigned* __restrict__ emax_u, float* __restrict__ sum_exp,
                             int E) {
  int e = blockIdx.x * blockDim.x + threadIdx.x;
  if (e >= E) return;
  int d = ei[E + e];
  float m  = ord2f(emax_u[d]);
  float ex = __expf(e_val[e] - m);
  e_val[e] = ex;   // in-place: now holds exp value
  unsafeAtomicAdd(&sum_exp[d], ex);
}

// ---------- Kernel 6: alpha * Wx_j scatter-add (16 threads per edge, float4 each) ----------
__global__ void scatter_msgs(const int* __restrict__ ei, const float* __restrict__ e_val,
                             const float* __restrict__ sum_exp, const float* __restrict__ Wx,
                             float* __restrict__ hout, int E) {
  int t = blockIdx.x * blockDim.x + threadIdx.x;
  int e = t >> 4;
  int q = t & 15;
  if (e >= E) return;
  int s = ei[e];
  int d = ei[E + e];
  float alpha = e_val[e] / (sum_exp[d] + 1e-16f);
  float4 w = ((const float4*)(Wx + (size_t)s * OUT_CH))[q];
  float* hp = hout + (size_t)d * OUT_CH + q * 4;
  unsafeAtomicAdd(hp + 0, alpha * w.x);
  unsafeAtomicAdd(hp + 1, alpha * w.y);
  unsafeAtomicAdd(hp + 2, alpha * w.z);
  unsafeAtomicAdd(hp + 3, alpha * w.w);
}

// ---------- Kernel 7: ELU in place ----------
__global__ void elu_inplace(float* __restrict__ h, int total) {
  int i = blockIdx.x * blockDim.x + threadIdx.x;
  if (i >= total) return;
  float v = h[i];
  h[i] = (v > 0.0f) ? v : (__expf(v) - 1.0f);
}

extern "C" void kernel_launch(void* const* d_in, const int* in_sizes, int n_in,
                              void* d_out, int out_size, void* d_ws, size_t ws_size,
                              hipStream_t stream) {
  const float* x       = (const float*)d_in[0];
  const int*   ei      = (const int*)d_in[1];     // [2, E]: row0 = src, row1 = dst
  const float* W       = (const float*)d_in[2];
  const float* att_src = (const float*)d_in[3];
  const float* att_dst = (const float*)d_in[4];
  float* hout = (float*)d_out;

  const int nNodes = in_sizes[0] / IN_CH;
  const int E      = in_sizes[1] / 2;
  const int total  = nNodes * OUT_CH;

  // Workspace layout
  char* ws = (char*)d_ws;
  float*    Wx      = (float*)ws;    ws += (size_t)nNodes * OUT_CH * sizeof(float);
  float*    s_src   = (float*)ws;    ws += (size_t)nNodes * sizeof(float);
  float*    s_dst   = (float*)ws;    ws += (size_t)nNodes * sizeof(float);
  float*    e_val   = (float*)ws;    ws += (size_t)E * sizeof(float);
  unsigned* emax_u  = (unsigned*)ws; ws += (size_t)nNodes * sizeof(unsigned);
  float*    sum_exp = (float*)ws;    ws += (size_t)nNodes * sizeof(float);

  const int TPB = 256;

  // 1) GEMM: one wave per 16x16 tile, 8 waves per block
  int numMTiles = (nNodes + 15) / 16;
  int waves     = numMTiles * 4;
  int gblocks   = (waves + 7) / 8;
  gemm_xW<<<gblocks, TPB, 0, stream>>>(x, W, Wx, nNodes, numMTiles);

  // 2) init output + segment buffers
  init_buffers<<<(total + TPB - 1) / TPB, TPB, 0, stream>>>(hout, emax_u, sum_exp, nNodes, total);

  // 3) per-node attention scalars
  node_scores<<<(nNodes + TPB - 1) / TPB, TPB, 0, stream>>>(Wx, att_src, att_dst, s_src, s_dst, nNodes);

  // 4) edge scores + segment max
  edge_scores_max<<<(E + TPB - 1) / TPB, TPB, 0, stream>>>(ei, s_src, s_dst, e_val, emax_u, E);

  // 5) exp + segment sum
  edge_exp_sum<<<(E + TPB - 1) / TPB, TPB, 0, stream>>>(ei, e_val, emax_u, sum_exp, E);

  // 6) weighted scatter (16 threads/edge)
  long long sthreads = (long long)E * 16;
  scatter_msgs<<<(int)((sthreads + TPB - 1) / TPB), TPB, 0, stream>>>(ei, e_val, sum_exp, Wx, hout, E);

  // 7) ELU
  elu_inplace<<<(total + TPB - 1) / TPB, TPB, 0, stream>>>(hout, total);
}